// PopNet_13159779795202
// MI455X (gfx1250) — compile-verified
//
#include <hip/hip_runtime.h>

typedef __attribute__((ext_vector_type(2))) float v2f;
typedef __attribute__((ext_vector_type(8))) float v8f;

// Problem constants (match reference)
constexpr int NB     = 131072;   // batch
constexpr int NSTEPS = 25;
constexpr int NH     = 64;       // hidden units
constexpr int NO     = 3;        // output units
constexpr int NF     = 4;        // raw features
constexpr int NBINS  = 15;
constexpr int NIN    = 60;       // NF * NBINS
constexpr float BETA = 0.8f;
constexpr float THR  = 1.0f;

// Output layout (flat, in reference return order):
//   spk2_rec : [25, B, 3]   offset 0
//   spk1_rec : [25, B, 64]  offset 25*B*3
//   x_pop    : [B, 60]      offset 25*B*(3+64)
constexpr size_t SPK2_ELEMS = (size_t)NSTEPS * NB * NO;
constexpr size_t SPK1_ELEMS = (size_t)NSTEPS * NB * NH;

__device__ __forceinline__ int bin_index(float xv) {
    xv = fminf(fmaxf(xv, 0.0f), 1.0f);
    int i = (int)(xv * (float)(NBINS - 1));   // truncation, xv >= 0
    return i > (NBINS - 1) ? (NBINS - 1) : i;
}

// One wave32 processes a 16-batch tile through all 25 SNN steps.
// mem1/cur1 live in the WMMA f32 16x16x4 A-matrix layout:
//   lane L -> batch (L & 15); reg pair c -> units 4c+{0,1} (lanes 0-15)
//                                      or units 4c+{2,3} (lanes 16-31).
// Layer-2 matmul (spk1 @ W2.T) is done with 16 chained
// V_WMMA_F32_16X16X4_F32 accumulating cur2 (f32 end-to-end).
__global__ __launch_bounds__(256)
void popnet_snn_kernel(const float* __restrict__ x,
                       const float* __restrict__ W1,
                       const float* __restrict__ b1,
                       const float* __restrict__ W2,
                       const float* __restrict__ b2,
                       float* __restrict__ out)
{
    __shared__ float lW1[NH * NIN];   // [64,60] row-major (out, in)
    __shared__ float lb1[NH];
    __shared__ float lb2[16];         // b2 padded to 16 (lane%16 indexed)
    __shared__ v2f   lB[16 * 32];     // B operand, pre-swizzled: [chunk][lane]

    const int tid = threadIdx.x;

    for (int i = tid; i < NH * NIN; i += 256) lW1[i] = W1[i];
    if (tid < NH) lb1[tid] = b1[tid];
    if (tid < 16) lb2[tid] = (tid < NO) ? b2[tid] : 0.0f;
    // Build the 4x16 f32 B operand per K-chunk c:
    //   B[k][n] = W2[n, 4c+k] for n<3 else 0
    //   layout: VGPR0 = rows K = {0 | 2} (lane halves), VGPR1 = rows K = {1 | 3}
    for (int t = tid; t < 16 * 32; t += 256) {
        const int c  = t >> 5;
        const int L  = t & 31;
        const int n  = L & 15;
        const int k0 = (L < 16) ? 0 : 2;
        v2f bv;
        bv.x = (n < NO) ? W2[n * NH + 4 * c + k0]     : 0.0f;
        bv.y = (n < NO) ? W2[n * NH + 4 * c + k0 + 1] : 0.0f;
        lB[t] = bv;
    }
    __syncthreads();

    const int lane  = tid & 31;
    const int wave  = tid >> 5;
    const int b0    = blockIdx.x * 128 + wave * 16;  // 8 waves * 16 batch
    const int mrow  = lane & 15;
    const int myb   = b0 + mrow;                     // this lane's batch (A layout)
    const int khalf = (lane < 16) ? 0 : 2;           // K sub-offset for this lane half

    // cur1[b,u] = b1[u] + sum_f W1[u, f*15 + idx_f]  (one-hot gather, constant in time)
    const float4 xb = reinterpret_cast<const float4*>(x)[myb];
    const int c0 = bin_index(xb.x);
    const int c1 = NBINS     + bin_index(xb.y);
    const int c2 = 2 * NBINS + bin_index(xb.z);
    const int c3 = 3 * NBINS + bin_index(xb.w);

    float cur1[32], mem1[32];
#pragma unroll
    for (int c = 0; c < 16; c++) {
#pragma unroll
        for (int j = 0; j < 2; j++) {
            const int u = 4 * c + khalf + j;
            const float* wr = &lW1[u * NIN];
            cur1[2 * c + j] = lb1[u] + wr[c0] + wr[c1] + wr[c2] + wr[c3];
            mem1[2 * c + j] = 0.0f;
        }
    }

    // C-layout b2 broadcast: element (M,N) needs b2[N]; N = lane%16
    const float binit = lb2[lane & 15];
    v8f cinit, mem2;
#pragma unroll
    for (int r = 0; r < 8; r++) { cinit[r] = binit; mem2[r] = 0.0f; }

    float* spk2_out = out;
    float* spk1_out = out + SPK2_ELEMS;

    const int n_col  = lane & 15;                  // C-layout column (output unit)
    const int mbase2 = b0 + ((lane >> 4) << 3);    // C-layout row base for this lane half

    for (int s = 0; s < NSTEPS; s++) {
        // Layer 1: reset(detached) -> leak+integrate -> fire
        float spk1[32];
#pragma unroll
        for (int v = 0; v < 32; v++) {
            float m   = mem1[v];
            float rst = (m > THR) ? THR : 0.0f;
            m = BETA * m + cur1[v] - rst;
            mem1[v] = m;
            spk1[v] = (m > THR) ? 1.0f : 0.0f;
        }

        // cur2 = spk1 @ W2.T + b2 via 16 chained f32 WMMAs (K = 64 = 16 * 4)
        v8f acc = cinit;
#pragma unroll
        for (int c = 0; c < 16; c++) {
            v2f a; a.x = spk1[2 * c]; a.y = spk1[2 * c + 1];
            v2f b = lB[(c << 5) + lane];
            acc = __builtin_amdgcn_wmma_f32_16x16x4_f32(
                false, a, false, b, (short)0, acc, false, false);
        }

        // Stream spk1_rec[s, myb, :] — unit pairs are contiguous -> b64 stores
        {
            const size_t rowbase = ((size_t)s * NB + (size_t)myb) * NH + khalf;
#pragma unroll
            for (int c = 0; c < 16; c++) {
                v2f sv; sv.x = spk1[2 * c]; sv.y = spk1[2 * c + 1];
                *reinterpret_cast<v2f*>(spk1_out + rowbase + 4 * c) = sv;
            }
        }

        // Layer 2: membrane update + fire in the C/D layout
        float spk2v[8];
#pragma unroll
        for (int r = 0; r < 8; r++) {
            float m   = mem2[r];
            float rst = (m > THR) ? THR : 0.0f;
            m = BETA * m + acc[r] - rst;
            mem2[r]  = m;
            spk2v[r] = (m > THR) ? 1.0f : 0.0f;
        }
        if (n_col < NO) {
            const size_t base = ((size_t)s * NB + (size_t)mbase2) * NO + n_col;
#pragma unroll
            for (int r = 0; r < 8; r++)
                spk2_out[base + (size_t)r * NO] = spk2v[r];
        }
        // reconverged here -> EXEC all-ones before next step's WMMAs
    }
}

// x_pop [B,60]: one thread per output element, fully coalesced stores.
__global__ __launch_bounds__(256)
void pop_encode_kernel(const float* __restrict__ x, float* __restrict__ xpop)
{
    const int i = blockIdx.x * blockDim.x + threadIdx.x;  // NB*NIN total, exact multiple
    const int b   = i / NIN;
    const int j   = i - b * NIN;
    const int f   = j / NBINS;
    const int bin = j - f * NBINS;
    const int idx = bin_index(x[b * NF + f]);
    xpop[i] = (bin == idx) ? 1.0f : 0.0f;
}

extern "C" void kernel_launch(void* const* d_in, const int* in_sizes, int n_in,
                              void* d_out, int out_size, void* d_ws, size_t ws_size,
                              hipStream_t stream)
{
    const float* x  = (const float*)d_in[0];
    const float* W1 = (const float*)d_in[1];
    const float* b1 = (const float*)d_in[2];
    const float* W2 = (const float*)d_in[3];
    const float* b2 = (const float*)d_in[4];
    float* out = (float*)d_out;

    // SNN: 1024 blocks * 256 threads; each wave32 owns a 16-batch tile
    popnet_snn_kernel<<<NB / 128, 256, 0, stream>>>(x, W1, b1, W2, b2, out);

    // x_pop: 131072*60 elements
    pop_encode_kernel<<<(NB * NIN) / 256, 256, 0, stream>>>(
        x, out + SPK2_ELEMS + SPK1_ELEMS);
}